// Net1_51221779972598
// MI455X (gfx1250) — compile-verified
//
#include <hip/hip_runtime.h>

typedef __attribute__((ext_vector_type(2))) float v2f;
typedef __attribute__((ext_vector_type(8))) float v8f;

#define WMMA_F32(a, b, c) \
  __builtin_amdgcn_wmma_f32_16x16x4_f32(false, (a), false, (b), (short)0, (c), false, false)

constexpr int   kB        = 4096;
constexpr int   kT        = 8;
constexpr int   kC        = 2;
constexpr int   kREL      = 64;
constexpr int   kEMB      = 128;
constexpr int   kGRAPH    = 256;
constexpr int   kACT      = 4096;
constexpr int   kFEAT     = kGRAPH + kEMB;   // 384
constexpr float kFLOATMIN = -3.4e38f;

// ---------------------------------------------------------------------------
// Kernel 1: RWt[g][m] = (rel_emb[64,128] @ graph_weight[128,256]) transposed.
// Transposed store makes k_graph's B fragments contiguous (b64 loads).
// One wave per 16x16 tile; 4 m-tiles x 16 n-tiles = 64 blocks. One-shot, tiny.
// ---------------------------------------------------------------------------
__global__ void __launch_bounds__(32) k_relgw(const float* __restrict__ rel_emb,
                                              const float* __restrict__ gw,
                                              float* __restrict__ RWt) {
  const int lane = threadIdx.x & 31;
  const int ln = lane & 15, lh = lane >> 4;
  const int mt = blockIdx.x & 3, nt = blockIdx.x >> 2;
  v8f acc = {};
  for (int k0 = 0; k0 < kEMB; k0 += 4) {
    const int k = k0 + lh * 2;
    v2f a, b;
    a.x = rel_emb[(mt * 16 + ln) * kEMB + k];
    a.y = rel_emb[(mt * 16 + ln) * kEMB + k + 1];
    b.x = gw[k * kGRAPH + nt * 16 + ln];
    b.y = gw[(k + 1) * kGRAPH + nt * 16 + ln];
    acc = WMMA_F32(a, b, acc);
  }
  for (int r = 0; r < 8; ++r)
    RWt[(nt * 16 + ln) * kREL + mt * 16 + r + lh * 8] = acc[r];
}

// ---------------------------------------------------------------------------
// Kernel 2: per batch b: OG = link[b](64x64) @ RW(64x256);
// g_pool[b][g] = sum_n relu(OG[n][g] + bias[g]) -> feat[b][0:256]
// One block (4 waves) per batch: link[b] staged once in LDS, each wave caches
// its full A-strip (K=64) in 16 v2f registers and sweeps all 16 n-tiles, so
// per WMMA there is exactly one global b64 load (L2-resident RWt).
// ---------------------------------------------------------------------------
__global__ void __launch_bounds__(128) k_graph(const float* __restrict__ link,
                                               const float* __restrict__ RWt,
                                               const float* __restrict__ bias,
                                               float* __restrict__ feat) {
  __shared__ float Asl[kREL * kREL];   // 16 KB: link[b]
  __shared__ float accg[kGRAPH];       // 1 KB: pooled columns
  const int b = blockIdx.x;
  const int wave = threadIdx.x >> 5;   // m-tile 0..3
  const int lane = threadIdx.x & 31;
  const int ln = lane & 15, lh = lane >> 4;

  {  // stage link[b] (4096 floats) with b128 copies
    const float4* src = (const float4*)(link + (size_t)b * kREL * kREL);
    float4* dst = (float4*)Asl;
    for (int i = threadIdx.x; i < kREL * kREL / 4; i += 128) dst[i] = src[i];
  }
  for (int i = threadIdx.x; i < kGRAPH; i += 128) accg[i] = 0.f;
  __syncthreads();

  // cache this wave's A fragments for the whole K range
  v2f af[kREL / 4];
  for (int k0 = 0; k0 < kREL; k0 += 4) {
    const int k = k0 + lh * 2;
    af[k0 >> 2].x = Asl[(wave * 16 + ln) * kREL + k];
    af[k0 >> 2].y = Asl[(wave * 16 + ln) * kREL + k + 1];
  }

  for (int nt = 0; nt < kGRAPH / 16; ++nt) {
    const float* Brow = RWt + (size_t)(nt * 16 + ln) * kREL;
    v8f acc = {};
    for (int k0 = 0; k0 < kREL; k0 += 4) {
      const int k = k0 + lh * 2;
      v2f bb;
      bb.x = Brow[k];
      bb.y = Brow[k + 1];
      acc = WMMA_F32(af[k0 >> 2], bb, acc);
    }
    const float bv = bias[nt * 16 + ln];
    float ps = 0.f;
    for (int r = 0; r < 8; ++r) ps += fmaxf(acc[r] + bv, 0.f);
    ps += __shfl_xor(ps, 16, 32);  // combine lane halves (same column)
    if (lane < 16) atomicAdd(&accg[nt * 16 + ln], ps);  // ds_add_f32
  }
  __syncthreads();
  for (int g = threadIdx.x; g < kGRAPH; g += 128)
    feat[(size_t)b * kFEAT + g] = accg[g];
}

// ---------------------------------------------------------------------------
// Kernel 3: per batch b: build q_emb (32x128) in LDS, K/Q/V = q_emb @ W^T via
// WMMA into LDS, then per-tree 4x4 softmax attention -> feat[b][256:384].
// Block = 256 threads (8 waves). LDS = 64 KB (acc aliases dead q buffer).
// ---------------------------------------------------------------------------
__global__ void __launch_bounds__(256) k_tree(const int* __restrict__ l_tab_id,
                                              const int* __restrict__ r_tab_id,
                                              const int* __restrict__ l_col_id,
                                              const int* __restrict__ r_col_id,
                                              const float* __restrict__ col_emb,
                                              const float* __restrict__ rel_emb,
                                              const float* __restrict__ Wk,
                                              const float* __restrict__ Wq,
                                              const float* __restrict__ Wv,
                                              float* __restrict__ feat) {
  __shared__ float smem[16384];      // 64 KB
  float* q_s   = smem;               // [32][128]
  float* kqv_s = smem + 4096;        // [3][32][128]: 0=K, 1=Q, 2=V
  float* acc_s = smem;               // aliases q_s once q is dead

  const int b = blockIdx.x;
  const int tid = threadIdx.x;
  const int wave = tid >> 5;
  const int lane = tid & 31;
  const int ln = lane & 15, lh = lane >> 4;

  // Gather q_emb rows: row = t*4 + i, i in {l_tab, l_col, r_col, r_tab}
  {
    const int row = tid >> 3;        // 0..31
    const int t = row >> 2, i = row & 3;
    const int d0 = (tid & 7) * 16;
    const float* src0;
    const float* src1 = nullptr;
    if (i == 0) {
      src0 = rel_emb + l_tab_id[b * kT + t] * kEMB;
    } else if (i == 3) {
      src0 = rel_emb + r_tab_id[b * kT + t] * kEMB;
    } else {
      const int* ids = (i == 1) ? l_col_id : r_col_id;
      src0 = col_emb + ids[(b * kT + t) * kC + 0] * kEMB;
      src1 = col_emb + ids[(b * kT + t) * kC + 1] * kEMB;
    }
    for (int d = d0; d < d0 + 16; ++d) {
      float v = src0[d];
      if (src1) v += src1[d];
      q_s[row * kEMB + d] = v;
    }
  }
  __syncthreads();

  // K,Q,V projections: 3 mats x (2 m-tiles x 8 n-tiles) = 48 tiles, 6/wave.
  // out[.., e] = sum_d q_emb[.., d] * W[e][d]  ->  B[k=d][n=e] = W[e*128+d]
  for (int tt = wave; tt < 48; tt += 8) {
    const int mat = tt >> 4, rem = tt & 15, mt = rem >> 3, nt = rem & 7;
    const float* W = (mat == 0) ? Wk : (mat == 1) ? Wq : Wv;
    v8f acc = {};
    for (int k0 = 0; k0 < kEMB; k0 += 4) {
      const int k = k0 + lh * 2;
      v2f a, bb;
      a.x = q_s[(mt * 16 + ln) * kEMB + k];
      a.y = q_s[(mt * 16 + ln) * kEMB + k + 1];
      bb.x = W[(nt * 16 + ln) * kEMB + k];
      bb.y = W[(nt * 16 + ln) * kEMB + k + 1];
      acc = WMMA_F32(a, bb, acc);
    }
    for (int r = 0; r < 8; ++r)
      kqv_s[mat * 4096 + (mt * 16 + r + lh * 8) * kEMB + nt * 16 + ln] = acc[r];
  }
  __syncthreads();
  if (tid < kEMB) acc_s[tid] = 0.f;
  __syncthreads();

  // Attention: wave handles tree t = wave. scores 4x4 on lanes 0..15.
  {
    const int t = wave;
    float w = 0.f;
    if (lane < 16) {
      const int i = lane >> 2, j = lane & 3;
      const float* qr = kqv_s + 1 * 4096 + (t * 4 + i) * kEMB;
      const float* kr = kqv_s + 0 * 4096 + (t * 4 + j) * kEMB;
      float s = 0.f;
      for (int d = 0; d < kEMB; ++d) s += qr[d] * kr[d];
      s *= (1.0f / kEMB);
      float m = s;
      m = fmaxf(m, __shfl_xor(m, 1, 32));
      m = fmaxf(m, __shfl_xor(m, 2, 32));
      const float e = __expf(s - m);
      float se = e;
      se += __shfl_xor(se, 1, 32);
      se += __shfl_xor(se, 2, 32);
      const float att = e / se;
      // column-sum over i (einsum collapses i): lanes j, j+4, j+8, j+12
      w = att;
      w += __shfl_xor(w, 4, 32);
      w += __shfl_xor(w, 8, 32);
    }
    const float w0 = __shfl(w, 0, 32);
    const float w1 = __shfl(w, 1, 32);
    const float w2 = __shfl(w, 2, 32);
    const float w3 = __shfl(w, 3, 32);
    const float* V = kqv_s + 2 * 4096 + (t * 4) * kEMB;
    for (int rep = 0; rep < 4; ++rep) {
      const int d = rep * 32 + lane;
      const float val = w0 * V[0 * kEMB + d] + w1 * V[1 * kEMB + d] +
                        w2 * V[2 * kEMB + d] + w3 * V[3 * kEMB + d];
      atomicAdd(&acc_s[d], val);   // ds_add_f32
    }
  }
  __syncthreads();
  if (tid < kEMB) feat[(size_t)b * kFEAT + kGRAPH + tid] = acc_s[tid];
}

// ---------------------------------------------------------------------------
// Kernel 4: out[b][a] = feat[b][:] . fc_w[a][:] + clip(log(mask[b][a]))
// grid (128 m-blocks of 32 rows, 32), 8 waves each owning one n-tile and
// computing TWO m-tiles that share every B fragment (1 global b64 : 2 WMMA).
// ---------------------------------------------------------------------------
__global__ void __launch_bounds__(256) k_fc(const float* __restrict__ feat,
                                            const float* __restrict__ fc_w,
                                            const float* __restrict__ mask,
                                            float* __restrict__ out) {
  __shared__ float Al[32 * kFEAT];  // 48 KB: two A m-tiles
  const int mb = blockIdx.x;        // rows mb*32 .. mb*32+31
  const int wave = threadIdx.x >> 5;
  const int lane = threadIdx.x & 31;
  const int ln = lane & 15, lh = lane >> 4;
  const int nt = blockIdx.y * 8 + wave;

  for (int i = threadIdx.x; i < 32 * kFEAT; i += 256)
    Al[i] = feat[(size_t)mb * 32 * kFEAT + i];
  __syncthreads();

  v8f acc0 = {}, acc1 = {};
  const float* Brow = fc_w + (size_t)(nt * 16 + ln) * kFEAT;
  for (int k0 = 0; k0 < kFEAT; k0 += 4) {
    const int k = k0 + lh * 2;
    v2f bb, a0, a1;
    bb.x = Brow[k];
    bb.y = Brow[k + 1];
    a0.x = Al[ln * kFEAT + k];
    a0.y = Al[ln * kFEAT + k + 1];
    a1.x = Al[(16 + ln) * kFEAT + k];
    a1.y = Al[(16 + ln) * kFEAT + k + 1];
    acc0 = WMMA_F32(a0, bb, acc0);
    acc1 = WMMA_F32(a1, bb, acc1);
  }

  const int n = nt * 16 + ln;
  for (int r = 0; r < 8; ++r) {
    const int m0 = mb * 32 + r + lh * 8;
    const int m1 = m0 + 16;
    const float mv0 = mask[(size_t)m0 * kACT + n];
    const float mv1 = mask[(size_t)m1 * kACT + n];
    out[(size_t)m0 * kACT + n] = acc0[r] + (mv0 > 0.f ? 0.f : kFLOATMIN);
    out[(size_t)m1 * kACT + n] = acc1[r] + (mv1 > 0.f ? 0.f : kFLOATMIN);
  }
}

// ---------------------------------------------------------------------------
extern "C" void kernel_launch(void* const* d_in, const int* in_sizes, int n_in,
                              void* d_out, int out_size, void* d_ws, size_t ws_size,
                              hipStream_t stream) {
  const float* link    = (const float*)d_in[0];
  const float* mask    = (const float*)d_in[1];
  const int*   l_tab   = (const int*)d_in[2];
  const int*   r_tab   = (const int*)d_in[3];
  const int*   l_col   = (const int*)d_in[4];
  const int*   r_col   = (const int*)d_in[5];
  const float* col_emb = (const float*)d_in[6];
  const float* rel_emb = (const float*)d_in[7];
  const float* gw      = (const float*)d_in[8];
  const float* gb      = (const float*)d_in[9];
  const float* Wk      = (const float*)d_in[10];
  const float* Wq      = (const float*)d_in[11];
  const float* Wv      = (const float*)d_in[12];
  const float* fc_w    = (const float*)d_in[13];
  float* out = (float*)d_out;

  float* RWt  = (float*)d_ws;            // 256*64 floats (64 KB), [g][m]
  float* feat = RWt + kREL * kGRAPH;     // 4096*384 floats (6.3 MB)

  k_relgw<<<dim3(64), dim3(32), 0, stream>>>(rel_emb, gw, RWt);
  k_graph<<<dim3(kB), dim3(128), 0, stream>>>(link, RWt, gb, feat);
  k_tree<<<dim3(kB), dim3(256), 0, stream>>>(l_tab, r_tab, l_col, r_col,
                                             col_emb, rel_emb, Wk, Wq, Wv, feat);
  k_fc<<<dim3(kB / 32, kACT / 16 / 8), dim3(256), 0, stream>>>(feat, fc_w, mask, out);
}